// CRD_57664230916482
// MI455X (gfx1250) — compile-verified
//
#include <hip/hip_runtime.h>
#include <hip/hip_bf16.h>

typedef __attribute__((ext_vector_type(2))) float v2f;
typedef __attribute__((ext_vector_type(8))) float v8f;

// ---------------------------------------------------------------------------
// 1) zero fill
// ---------------------------------------------------------------------------
__global__ void CRD_zero_f32(float* __restrict__ p, long long n) {
  long long i = (long long)blockIdx.x * blockDim.x + threadIdx.x;
  if (i < n) p[i] = 0.0f;
}

// ---------------------------------------------------------------------------
// 2) degree accumulation: deg_out[src[e]] += 1, deg_in[dst[e]] += 1
// ---------------------------------------------------------------------------
__global__ void CRD_edge_degree(const int* __restrict__ src,
                                const int* __restrict__ dst,
                                float* __restrict__ deg_out,
                                float* __restrict__ deg_in, int E) {
  int e = blockIdx.x * blockDim.x + threadIdx.x;
  if (e < E) {
    atomicAdd(&deg_out[src[e]], 1.0f);
    atomicAdd(&deg_in[dst[e]], 1.0f);
  }
}

// ---------------------------------------------------------------------------
// 3) in-place: deg -> clip(deg,1)^{-1/2}   (covers both arrays contiguously)
// ---------------------------------------------------------------------------
__global__ void CRD_make_norm(float* __restrict__ deg, int n) {
  int i = blockIdx.x * blockDim.x + threadIdx.x;
  if (i < n) {
    float d = deg[i];
    d = d < 1.0f ? 1.0f : d;
    deg[i] = rsqrtf(d);
  }
}

// ---------------------------------------------------------------------------
// 4) WMMA GEMM: x[m, 0:64] = (h[m,:]*norm_src[m]) @ W[256,64]
//    One wave per 16-row slab; 4 accumulator tiles (N = 0..63 in 16-col tiles).
//    V_WMMA_F32_16X16X4_F32, fp32 end-to-end (matches reference precision).
//    A 16x4 layout: lanes 0-15 -> M=0..15, K = {k, k+1}; lanes 16-31 -> K = {k+2, k+3}.
//    B 4x16 layout: VGPR0 lanes0-15 = row k (N), lanes16-31 = row k+2; VGPR1 = k+1 / k+3.
//    C/D 16x16:     VGPR r, lanes0-15 -> M=r, N=lane; lanes16-31 -> M=r+8.
// ---------------------------------------------------------------------------
__global__ void __launch_bounds__(128)
CRD_gemm_wmma(const float* __restrict__ h, const float* __restrict__ W,
              const float* __restrict__ norm_src, float* __restrict__ x,
              int n_nodes) {
  const int lane = threadIdx.x & 31;
  const int wave = threadIdx.x >> 5;
  const int slab = blockIdx.x * 4 + wave;     // 16-row slab
  if (slab * 16 >= n_nodes) return;           // wave-uniform guard (EXEC stays all-1)

  const int row0 = slab * 16;
  const int mrow = row0 + (lane & 15);        // A-matrix row for this lane
  const int koff = (lane < 16) ? 0 : 2;       // half-wave K offset
  const int ncol = lane & 15;                 // column within a 16-wide tile
  const float ns = norm_src[mrow];

  const float* hrow = h + (size_t)mrow * 256 + koff;

  v8f acc0 = {}, acc1 = {}, acc2 = {}, acc3 = {};

  for (int k = 0; k < 256; k += 4) {
    // A fragment: two consecutive K values from this lane's row, norm folded in
    v2f a;
    a.x = hrow[k]     * ns;
    a.y = hrow[k + 1] * ns;

    // B fragments for the 4 N-tiles: rows (k+koff) and (k+koff+1), coalesced in N
    const float* wb = W + (size_t)(k + koff) * 64 + ncol;
    v2f b0, b1, b2, b3;
    b0.x = wb[0];       b0.y = wb[64 + 0];
    b1.x = wb[16];      b1.y = wb[64 + 16];
    b2.x = wb[32];      b2.y = wb[64 + 32];
    b3.x = wb[48];      b3.y = wb[64 + 48];

    acc0 = __builtin_amdgcn_wmma_f32_16x16x4_f32(false, a, false, b0, (short)0, acc0, false, false);
    acc1 = __builtin_amdgcn_wmma_f32_16x16x4_f32(false, a, false, b1, (short)0, acc1, false, false);
    acc2 = __builtin_amdgcn_wmma_f32_16x16x4_f32(false, a, false, b2, (short)0, acc2, false, false);
    acc3 = __builtin_amdgcn_wmma_f32_16x16x4_f32(false, a, false, b3, (short)0, acc3, false, false);
  }

  // Store D tiles per C/D VGPR layout
  const int srow0 = row0 + ((lane >> 4) ? 8 : 0);
  float* xo = x + (size_t)srow0 * 64 + ncol;
#pragma unroll
  for (int r = 0; r < 8; ++r) {
    xo[(size_t)r * 64 + 0]  = acc0[r];
    xo[(size_t)r * 64 + 16] = acc1[r];
    xo[(size_t)r * 64 + 32] = acc2[r];
    xo[(size_t)r * 64 + 48] = acc3[r];
  }
}

// ---------------------------------------------------------------------------
// 5) scatter-add: agg[dst[e], :] += x[src[e], :]   (16 lanes/edge, float4 each)
// ---------------------------------------------------------------------------
__global__ void CRD_scatter_edges(const float* __restrict__ x,
                                  const int* __restrict__ src,
                                  const int* __restrict__ dst,
                                  float* __restrict__ agg, int E) {
  long long t = (long long)blockIdx.x * blockDim.x + threadIdx.x;
  int e = (int)(t >> 4);
  if (e >= E) return;
  int c = ((int)t & 15) * 4;
  const float4 v = *(const float4*)(x + (size_t)src[e] * 64 + c);
  float* o = agg + (size_t)dst[e] * 64 + c;
  atomicAdd(o + 0, v.x);
  atomicAdd(o + 1, v.y);
  atomicAdd(o + 2, v.z);
  atomicAdd(o + 3, v.w);
}

// ---------------------------------------------------------------------------
// 6) finalize in place: out = relu(out * norm_dst[node] + b[col])
// ---------------------------------------------------------------------------
__global__ void CRD_finalize(float* __restrict__ out,
                             const float* __restrict__ norm_dst,
                             const float* __restrict__ b, int total) {
  int i = blockIdx.x * blockDim.x + threadIdx.x;
  if (i < total) {
    int node = i >> 6;
    int col  = i & 63;
    float v = out[i] * norm_dst[node] + b[col];
    out[i] = v > 0.0f ? v : 0.0f;
  }
}

// ---------------------------------------------------------------------------
extern "C" void kernel_launch(void* const* d_in, const int* in_sizes, int n_in,
                              void* d_out, int out_size, void* d_ws, size_t ws_size,
                              hipStream_t stream) {
  const float* h   = (const float*)d_in[0];
  const int*   src = (const int*)d_in[1];
  const int*   dst = (const int*)d_in[2];
  const float* W   = (const float*)d_in[3];
  const float* b   = (const float*)d_in[4];

  const int N = in_sizes[0] / 256;   // 100000 nodes
  const int E = in_sizes[1];         // 1600000 edges

  float* norm_src = (float*)d_ws;          // [N]
  float* norm_dst = norm_src + N;          // [N]  (contiguous with norm_src)
  float* x        = norm_dst + N;          // [N, 64]
  float* out      = (float*)d_out;         // [N, 64] accumulator then result

  const int B256 = 256;

  // zero degree accumulators (2N) and output accumulator (N*64)
  {
    long long n = 2LL * N;
    CRD_zero_f32<<<(int)((n + B256 - 1) / B256), B256, 0, stream>>>(norm_src, n);
    long long m = (long long)N * 64;
    CRD_zero_f32<<<(int)((m + B256 - 1) / B256), B256, 0, stream>>>(out, m);
  }

  // degrees
  CRD_edge_degree<<<(E + B256 - 1) / B256, B256, 0, stream>>>(src, dst, norm_src, norm_dst, E);

  // deg -> clip(deg,1)^{-1/2}  (both arrays in one pass)
  CRD_make_norm<<<(2 * N + B256 - 1) / B256, B256, 0, stream>>>(norm_src, 2 * N);

  // WMMA GEMM: x = (h * norm_src) @ W
  {
    int slabs  = (N + 15) / 16;           // 6250
    int blocks = (slabs + 3) / 4;         // 4 waves (slabs) per 128-thread block
    CRD_gemm_wmma<<<blocks, 128, 0, stream>>>(h, W, norm_src, x, N);
  }

  // scatter-add messages into out
  {
    long long threads = (long long)E * 16;
    CRD_scatter_edges<<<(int)((threads + B256 - 1) / B256), B256, 0, stream>>>(x, src, dst, out, E);
  }

  // out = relu(out * norm_dst + b)
  CRD_finalize<<<(N * 64 + B256 - 1) / B256, B256, 0, stream>>>(out, norm_dst, b, N * 64);
}